// CpSae_44014824849572
// MI455X (gfx1250) — compile-verified
//
#include <hip/hip_runtime.h>
#include <hip/hip_bf16.h>

// ---------------------------------------------------------------------------
// CP-SAE fused loss for MI455X (gfx1250, wave32).
// Heavy stages use V_WMMA_F32_16X16X4_F32 (fp32 matrix pipe):
//   k1: [1024x65536] x [65536x64] split-K GEMM (z_mu | z_log_std) with
//       double-buffered async global->LDS staging (ASYNCcnt pipeline)
//   k4: per-(b,f) 32x32x32 CP matmul fused with reconstruction MSE
// Both stages are HBM-bound (~536 MB features traffic @ 23.3 TB/s ~= 23 us),
// so fp32 WMMA is the right precision: no bandwidth to win by narrowing.
// ---------------------------------------------------------------------------

#define NFEAT   65536       // 64*32*32
#define ZDIM    32
#define NBATCH  1024
#define NFREQ   64
#define NROI    32
#define NGROUP  16
#define KCHUNK  4096        // K per workgroup in k1
#define KTILE   128         // K staged in LDS per pipeline stage in k1

typedef float v2f __attribute__((ext_vector_type(2)));
typedef float v8f __attribute__((ext_vector_type(8)));

// Async global->LDS copy (GLOBAL_LOAD_ASYNC_TO_LDS_B128 + S_WAIT_ASYNCCNT).
// Guarded: falls back to synchronous float4 staging if the builtins are absent.
#if defined(__has_builtin)
#if __has_builtin(__builtin_amdgcn_global_load_async_to_lds_b128) && \
    __has_builtin(__builtin_amdgcn_s_wait_asynccnt)
#define USE_ASYNC_LDS 1
#endif
#endif
#ifndef USE_ASYNC_LDS
#define USE_ASYNC_LDS 0
#endif

// Builtin signature (from the compiler's own diagnostic):
//   param0 = 'int __attribute__((vector_size(16))) __device__ *'  (addrspace(1))
//   param1 = LDS-side pointer (addrspace(3)), same element type
typedef int v4i_vs __attribute__((vector_size(16)));
typedef __attribute__((address_space(1))) v4i_vs* gv4_ptr;   // global v4i*
typedef __attribute__((address_space(3))) v4i_vs* lv4_ptr;   // LDS v4i*

// Generic->global: AS1 pointer value equals the flat address.
__device__ __forceinline__ gv4_ptr as_global_v4(const void* p) {
  return (gv4_ptr)(unsigned long long)p;
}
// Generic->LDS: AS3 pointer value is the low 32 bits of the flat shared
// address (identical to LLVM's generic->local addrspacecast lowering).
__device__ __forceinline__ lv4_ptr as_lds_v4(void* p) {
  return (lv4_ptr)(unsigned int)(unsigned long long)p;
}

__device__ __forceinline__ v8f wmma4(v2f a, v2f b, v8f c) {
  // D = A(16x4) * B(4x16) + C(16x16), fp32. 8-arg form:
  // (neg_a, A, neg_b, B, c_mod, C, reuse_a, reuse_b)
  return __builtin_amdgcn_wmma_f32_16x16x4_f32(false, a, false, b,
                                               (short)0, c, false, false);
}

__device__ __forceinline__ float wave_reduce(float v) {
  #pragma unroll
  for (int off = 16; off > 0; off >>= 1) v += __shfl_xor(v, off, 32);
  return v;
}

__device__ __forceinline__ float softplusf(float x) {
  return (x > 20.0f) ? x : log1pf(__expf(x));
}

// ---------------------------------------------------------------------------
// k0: zero accumulators (ws is poisoned 0xAA by the harness, d_out too)
// ---------------------------------------------------------------------------
__global__ void k0_init(float* __restrict__ acc, float* __restrict__ out,
                        int nacc, int nout) {
  int i = blockIdx.x * blockDim.x + threadIdx.x;
  if (i < nacc) acc[i] = 0.0f;
  if (i < nout) out[i] = 0.0f;
}

// ---------------------------------------------------------------------------
// k1: split-K GEMM  acc[b, 0:32] += flat@W1 ; acc[b, 32:64] += flat@W2
// grid = (64 M-tiles, 16 K-chunks), 128 threads = 4 waves.
// Wave w owns one 16x16 N-tile: w0=W1[:,0:16] w1=W1[:,16:32] w2=W2[:,0:16] w3=W2[:,16:32]
// A tile (16 rows x 128 K) double-buffered in LDS; staged with the async
// global->LDS engine when available (overlaps HBM latency with WMMA issue).
// ---------------------------------------------------------------------------
__global__ __launch_bounds__(128) void k1_gemm(const float* __restrict__ features,
                                               const float* __restrict__ W1,
                                               const float* __restrict__ W2,
                                               float* __restrict__ acc) {
  __shared__ float As[2][16 * KTILE];   // 2 x 8 KB
  const int tid = threadIdx.x;
  const int w   = tid >> 5;
  const int l   = tid & 31;
  const int mbase = blockIdx.x * 16;
  const int kbase = blockIdx.y * KCHUNK;

  const float* Wsel = (w < 2) ? W1 : W2;
  const int ncol = (w & 1) * 16 + (l & 15);
  const int m  = l & 15;
  const int kh = (l >> 4) * 2;          // lane half selects K pair {0,1} / {2,3}

  // Each thread stages 4 x 16B of the 16x128 tile (whole tile = 128 thr x 4).
  int rows[4], cols[4];
  #pragma unroll
  for (int i = 0; i < 4; ++i) {
    int idx = tid + 128 * i;
    rows[i] = idx >> 5;
    cols[i] = (idx & 31) * 4;
  }

  auto issue_tile = [&](int buf, int kt0) {
    #pragma unroll
    for (int i = 0; i < 4; ++i) {
      const float* src =
          features + (size_t)(mbase + rows[i]) * NFEAT + kbase + kt0 + cols[i];
      float* dst = &As[buf][rows[i] * KTILE + cols[i]];
#if USE_ASYNC_LDS
      // 4 async b128 ops per wave per tile -> ASYNCcnt == 4 per stage
      __builtin_amdgcn_global_load_async_to_lds_b128(as_global_v4(src),
                                                     as_lds_v4(dst), 0, 0);
#else
      *(float4*)dst = *(const float4*)src;
      __builtin_prefetch(src + KTILE, 0, 1);   // global_prefetch_b8
#endif
    }
  };

  v8f c = {};
  int buf = 0;
  issue_tile(0, 0);                      // prologue: tile 0 in flight
  for (int kt0 = 0; kt0 < KCHUNK; kt0 += KTILE) {
    const bool more = (kt0 + KTILE) < KCHUNK;
    if (more) issue_tile(buf ^ 1, kt0 + KTILE);   // next tile in flight
#if USE_ASYNC_LDS
    if (more) __builtin_amdgcn_s_wait_asynccnt(4);  // retire current tile only
    else      __builtin_amdgcn_s_wait_asynccnt(0);
#endif
    __syncthreads();                     // current tile visible to all waves
    const float* A = As[buf];
    #pragma unroll 4
    for (int kt = 0; kt < KTILE; kt += 4) {
      v2f a;
      a.x = A[m * KTILE + kt + kh];
      a.y = A[m * KTILE + kt + kh + 1];
      size_t kg = (size_t)(kbase + kt0 + kt + kh);
      v2f bf;
      bf.x = Wsel[kg * ZDIM + ncol];     // W rows are 32 floats, L2-hot
      bf.y = Wsel[(kg + 1) * ZDIM + ncol];
      c = wmma4(a, bf, c);
    }
    __syncthreads();                     // done reading before buf is re-staged
    buf ^= 1;
  }
  // C layout: lane l, vgpr v -> row = v + 8*(l>=16), col = l&15
  #pragma unroll
  for (int v = 0; v < 8; ++v) {
    int row = mbase + v + 8 * (l >> 4);
    int col = w * 16 + (l & 15);
    atomicAdd(&acc[row * 64 + col], c[v]);   // global_atomic_add_f32 split-K merge
  }
}

// ---------------------------------------------------------------------------
// k2: per-sample stats. One wave per sample (wave32 == ZDIM, lane == z).
//   embed tail + bias -> mu, log_std -> sigma, KL, rsample, zs@lin_W+lin_b,
//   classification term -> atomic loss; stores softplus(zs) for k4.
// ---------------------------------------------------------------------------
__global__ __launch_bounds__(256) void k2_stats(
    const float* __restrict__ acc, const float* __restrict__ b1,
    const float* __restrict__ b2, const float* __restrict__ W1,
    const float* __restrict__ W2, const float* __restrict__ gemb,
    const int* __restrict__ groups, const int* __restrict__ labels,
    const float* __restrict__ weights, const float* __restrict__ noise,
    const float* __restrict__ linW, const float* __restrict__ linb,
    const float* __restrict__ logit_bias, float* __restrict__ sz,
    float* __restrict__ out) {
  __shared__ float zbuf[8][32];
  __shared__ float zs2[8][32];
  const int w = threadIdx.x >> 5;
  const int l = threadIdx.x & 31;
  const int b = blockIdx.x * 8 + w;

  const int g = groups[b];
  const float e0 = gemb[g * 2 + 0];
  const float e1 = gemb[g * 2 + 1];

  float mu = acc[b * 64 + l] + b1[l]
           + e0 * W1[(size_t)NFEAT * ZDIM + l]
           + e1 * W1[(size_t)(NFEAT + 1) * ZDIM + l];
  float ls = acc[b * 64 + 32 + l] + b2[l]
           + e0 * W2[(size_t)NFEAT * ZDIM + l]
           + e1 * W2[(size_t)(NFEAT + 1) * ZDIM + l];

  float sigma = 1e-6f + __expf(ls);
  float kld = wave_reduce(-__logf(sigma) + 0.5f * (sigma * sigma + mu * mu - 1.0f));

  zbuf[w][l] = mu + sigma * noise[b * ZDIM + l];   // rsample
  __syncthreads();
  float zj = linb[l];
  #pragma unroll 8
  for (int i = 0; i < ZDIM; ++i) zj += zbuf[w][i] * linW[i * ZDIM + l];
  zs2[w][l] = zj;
  sz[b * ZDIM + l] = softplusf(zj);
  __syncthreads();

  if (l == 0) {
    float lg[4];
    lg[0] = zs2[w][0] + logit_bias[0];
    lg[1] = zs2[w][1] + logit_bias[1];
    lg[2] = zs2[w][2] + logit_bias[2];
    lg[3] = 1.0f + logit_bias[3];
    float mx = fmaxf(fmaxf(lg[0], lg[1]), fmaxf(lg[2], lg[3]));
    float se = __expf(lg[0] - mx) + __expf(lg[1] - mx) +
               __expf(lg[2] - mx) + __expf(lg[3] - mx);
    float logp = lg[labels[b]] - mx - __logf(se);
    atomicAdd(out, (-weights[b] * logp + kld) * (1.0f / (float)NBATCH));
  }
}

// ---------------------------------------------------------------------------
// k3: softplus factors + group variance losses (ddof=1 over 16 groups).
// 4096 threads: 2048 (z,f) freq positions + 2x1024 (z,r) roi positions.
// ---------------------------------------------------------------------------
__global__ void k3_factors(const float* __restrict__ ff,
                           const float* __restrict__ r1f,
                           const float* __restrict__ r2f,
                           float* __restrict__ fsp, float* __restrict__ r1sp,
                           float* __restrict__ r2sp, float* __restrict__ out) {
  int p = blockIdx.x * blockDim.x + threadIdx.x;
  if (p < 2048) {                       // freq: (z, f)
    int z = p >> 6, f = p & 63;
    float s = 0.0f, s2 = 0.0f;
    #pragma unroll 4
    for (int g = 0; g < NGROUP; ++g) {
      float v = softplusf(ff[(g * ZDIM + z) * NFREQ + f]);
      fsp[(g * ZDIM + z) * NFREQ + f] = v;
      s += v; s2 += v * v;
    }
    float var = (s2 - s * s * (1.0f / 16.0f)) * (1.0f / 15.0f);
    atomicAdd(out, var * (1.0f / (float)NFREQ));
  } else if (p < 4096) {                // roi: (z, r) for roi1 then roi2
    int idx = p - 2048;
    const float* src = (idx < 1024) ? r1f : r2f;
    float* dst = (idx < 1024) ? r1sp : r2sp;
    int q = idx & 1023;
    int z = q >> 5, r = q & 31;
    float s = 0.0f, s2 = 0.0f;
    #pragma unroll 4
    for (int g = 0; g < NGROUP; ++g) {
      float v = softplusf(src[(g * ZDIM + z) * NROI + r]);
      dst[(g * ZDIM + z) * NROI + r] = v;
      s += v; s2 += v * v;
    }
    float var = (s2 - s * s * (1.0f / 16.0f)) * (1.0f / 15.0f);
    atomicAdd(out, var * (1.0f / (float)NROI));
  }
}

// ---------------------------------------------------------------------------
// k4: CP volume + reconstruction MSE, fused. One WG per sample (1024 WGs,
// 8 waves). Per (b,f): vol[r,s] = sum_z (t[z]*R1[z,r]) * R2[z,s],
// t[z] = softplus(zs)[z]*F[z,f]  ->  32x32x32 matmul = 4 WMMA C-tiles x 8 K.
// R1/R2 fragments are f-invariant -> hoisted into registers once per wave;
// the per-f loop is only: t from LDS, 4 VALU mults, 4 WMMAs per K-step.
// volume never touches memory; features re-read once for the MSE.
// ---------------------------------------------------------------------------
__global__ __launch_bounds__(256) void k4_volume(
    const float* __restrict__ features, const int* __restrict__ groups,
    const float* __restrict__ sz, const float* __restrict__ fsp,
    const float* __restrict__ r1sp, const float* __restrict__ r2sp,
    float* __restrict__ out) {
  __shared__ float R1[ZDIM * NROI];     // [z][r]  4 KB
  __shared__ float R2[ZDIM * NROI];     // [z][s]  4 KB
  __shared__ float FQ[ZDIM * NFREQ];    // [z][f]  8 KB
  __shared__ float SZ[ZDIM];
  __shared__ float T[8][ZDIM];          // per-wave t[z]
  __shared__ float red[8];

  const int b = blockIdx.x;
  const int tid = threadIdx.x;
  const int w = tid >> 5;
  const int l = tid & 31;
  const int g = groups[b];

  for (int i = tid; i < ZDIM * NROI; i += 256) {
    R1[i] = r1sp[g * ZDIM * NROI + i];
    R2[i] = r2sp[g * ZDIM * NROI + i];
  }
  for (int i = tid; i < ZDIM * NFREQ; i += 256) FQ[i] = fsp[g * ZDIM * NFREQ + i];
  if (tid < ZDIM) SZ[tid] = sz[b * ZDIM + tid];
  __syncthreads();

  const int m  = l & 15;
  const int kh = (l >> 4) * 2;

  // Hoist f-invariant fragments: R1 rows (A-side scalars) and R2 B-fragments.
  v2f r1lo[8], r1hi[8], b0v[8], b1v[8];
  #pragma unroll
  for (int zc = 0; zc < 8; ++zc) {
    int za = zc * 4 + kh, zb = za + 1;
    r1lo[zc].x = R1[za * NROI + m];       r1lo[zc].y = R1[zb * NROI + m];
    r1hi[zc].x = R1[za * NROI + m + 16];  r1hi[zc].y = R1[zb * NROI + m + 16];
    b0v[zc].x  = R2[za * NROI + m];       b0v[zc].y  = R2[zb * NROI + m];
    b1v[zc].x  = R2[za * NROI + m + 16];  b1v[zc].y  = R2[zb * NROI + m + 16];
  }

  const float* feat = features + (size_t)b * NFEAT;
  float se = 0.0f;

  for (int fi = 0; fi < 8; ++fi) {
    const int f = w * 8 + fi;
    T[w][l] = SZ[l] * FQ[l * NFREQ + f];
    __syncthreads();

    v8f c00 = {}, c01 = {}, c10 = {}, c11 = {};
    #pragma unroll
    for (int zc = 0; zc < 8; ++zc) {
      int za = zc * 4 + kh, zb = za + 1;
      float ta = T[w][za], tb = T[w][zb];
      v2f a0 = {ta * r1lo[zc].x, tb * r1lo[zc].y};
      v2f a1 = {ta * r1hi[zc].x, tb * r1hi[zc].y};
      c00 = wmma4(a0, b0v[zc], c00);
      c01 = wmma4(a0, b1v[zc], c01);
      c10 = wmma4(a1, b0v[zc], c10);
      c11 = wmma4(a1, b1v[zc], c11);
    }

    const float* fp = feat + f * (NROI * NROI);
    #pragma unroll
    for (int v = 0; v < 8; ++v) {
      int r0 = v + 8 * (l >> 4);
      int s0 = l & 15;
      float d;
      d = fp[r0 * NROI + s0]             - c00[v]; se += d * d;
      d = fp[r0 * NROI + s0 + 16]        - c01[v]; se += d * d;
      d = fp[(r0 + 16) * NROI + s0]      - c10[v]; se += d * d;
      d = fp[(r0 + 16) * NROI + s0 + 16] - c11[v]; se += d * d;
    }
    __syncthreads();                     // before T is rewritten next iter
  }

  float wsum = wave_reduce(se);
  if (l == 0) red[w] = wsum;
  __syncthreads();
  if (tid == 0) {
    float tot = 0.0f;
    #pragma unroll
    for (int i = 0; i < 8; ++i) tot += red[i];
    // REG_STRENGTH=1: mean over 65536 feats, mean over 1024 batch
    atomicAdd(out, tot * (1.0f / ((float)NFEAT * (float)NBATCH)));
  }
}

// ---------------------------------------------------------------------------
extern "C" void kernel_launch(void* const* d_in, const int* in_sizes, int n_in,
                              void* d_out, int out_size, void* d_ws, size_t ws_size,
                              hipStream_t stream) {
  const float* features = (const float*)d_in[0];
  const int*   labels   = (const int*)d_in[1];
  const int*   groups   = (const int*)d_in[2];
  const float* weights  = (const float*)d_in[3];
  const float* noise    = (const float*)d_in[4];
  const float* gemb     = (const float*)d_in[5];
  const float* W1       = (const float*)d_in[6];
  const float* b1       = (const float*)d_in[7];
  const float* W2       = (const float*)d_in[8];
  const float* b2       = (const float*)d_in[9];
  const float* ff       = (const float*)d_in[10];
  const float* r1f      = (const float*)d_in[11];
  const float* r2f      = (const float*)d_in[12];
  const float* linW     = (const float*)d_in[13];
  const float* linb     = (const float*)d_in[14];
  const float* lbias    = (const float*)d_in[15];

  float* ws   = (float*)d_ws;
  float* acc  = ws;               // 1024*64      = 65536 floats (mu|log_std accum)
  float* szb  = ws + 65536;       // 1024*32      = 32768 (softplus(zs))
  float* fsp  = ws + 98304;       // 16*32*64     = 32768
  float* r1sp = ws + 131072;      // 16*32*32     = 16384
  float* r2sp = ws + 147456;      // 16*32*32     = 16384
  float* out  = (float*)d_out;

  k0_init<<<(65536 + 255) / 256, 256, 0, stream>>>(acc, out, 65536, out_size);
  k1_gemm<<<dim3(64, NFEAT / KCHUNK), 128, 0, stream>>>(features, W1, W2, acc);
  k2_stats<<<NBATCH / 8, 256, 0, stream>>>(acc, b1, b2, W1, W2, gemb, groups,
                                           labels, weights, noise, linW, linb,
                                           lbias, szb, out);
  k3_factors<<<16, 256, 0, stream>>>(ff, r1f, r2f, fsp, r1sp, r2sp, out);
  k4_volume<<<NBATCH, 256, 0, stream>>>(features, groups, szb, fsp, r1sp, r2sp,
                                        out);
}